// EnhancedCausalModel_51866025067017
// MI455X (gfx1250) — compile-verified
//
#include <hip/hip_runtime.h>
#include <hip/hip_bf16.h>
#include <math.h>

// Problem constants (match reference)
#define NA   16    // N_AGENTS
#define DOBS 256   // OBS_DIM
#define AD   32    // ACT_DIM
#define BB   16    // B
#define TT   64    // T
#define SS   16    // ADAPTIVE (MC samples)
#define H1D  128
#define H2D  64
#define BT   (BB*TT)       // 1024
#define NVAR 18            // p_with, p_zero, 16 cf samples
#define INV_S1 (1.0f/17.0f)

// Padded LDS row strides (stride mod 64 banks == 4 -> conflict-free A reads)
#define H1P  132
#define H2P  68
#define OBSP 260

// LDS layout (float offsets)
#define OFF_HOBS 0                      // 16*128  = 2048
#define OFF_H1   2048                   // 16*132  = 2112
#define OFF_H2   4160                   // 16*68   = 1088
#define OFF_PW   5248                   // 16*32   = 512
#define OFF_PQ   5760                   // 16*32   = 512
#define OFF_W1A  6272                   // 32*128  = 4096
#define OFF_W2   10368                  // 128*64  = 8192  (aliased by s_obs in step 1)
#define OFF_W3   18560                  // 64*32   = 2048
#define LDS_FLOATS 20608                // 82432 bytes

typedef float v2f __attribute__((ext_vector_type(2)));
typedef float v8f __attribute__((ext_vector_type(8)));

__device__ __forceinline__ v8f wmma4(v2f a, v2f b, v8f c) {
  // D(16x16,f32) = A(16x4,f32) x B(4x16,f32) + C  -- v_wmma_f32_16x16x4_f32
  return __builtin_amdgcn_wmma_f32_16x16x4_f32(
      /*neg_a=*/false, a, /*neg_b=*/false, b,
      /*c_mod=*/(short)0, c, /*reuse_a=*/false, /*reuse_b=*/false);
}

__device__ __forceinline__ float wave_max(float v) {
  #pragma unroll
  for (int off = 16; off; off >>= 1) v = fmaxf(v, __shfl_xor(v, off, 32));
  return v;
}
__device__ __forceinline__ float wave_sum(float v) {
  #pragma unroll
  for (int off = 16; off; off >>= 1) v += __shfl_xor(v, off, 32);
  return v;
}

__global__ __launch_bounds__(128)
void coma_influence_kernel(const float* __restrict__ obs,   // [B,T,N,DOBS]
                           const float* __restrict__ act,   // [B,T,N,AD]
                           const float* __restrict__ cf,    // [N,S,B,T,AD]
                           const float* __restrict__ W1,    // [288,128]
                           const float* __restrict__ b1,    // [128]
                           const float* __restrict__ W2,    // [128,64]
                           const float* __restrict__ b2,    // [64]
                           const float* __restrict__ W3,    // [64,32]
                           const float* __restrict__ b3,    // [32]
                           float* __restrict__ out)         // [B,T,N]
{
  extern __shared__ float smem[];
  float* s_hobs = smem + OFF_HOBS;
  float* s_h1   = smem + OFF_H1;
  float* s_h2   = smem + OFF_H2;
  float* s_pw   = smem + OFF_PW;
  float* s_pq   = smem + OFF_PQ;
  float* s_w1a  = smem + OFF_W1A;
  float* s_w2   = smem + OFF_W2;
  float* s_w3   = smem + OFF_W3;
  float* s_obs  = smem + OFF_W2;   // 16*260 = 4160 floats, aliases W2 region

  const int tid  = threadIdx.x;
  const int lane = tid & 31;
  const int wid  = tid >> 5;             // 4 waves per block
  const int g    = blockIdx.x;           // 1024 tiles of 16 rows
  const int r0   = g * 16;               // flat row = n*1024 + bt
  const int n    = r0 >> 10;             // agent id (16 rows never cross agents)
  const int bt0  = r0 & 1023;

  // WMMA fragment coordinates (wave32, 16x16x4 f32):
  //   A: row = lane&15, holds K = kb, kb+1 where kb = 2*(lane>>4)
  //   B: col = lane&15, holds K = kb, kb+1
  //   C/D: col = lane&15, row = v + 8*(lane>>4) for VGPR v in 0..7
  const int col = lane & 15;
  const int hi  = lane >> 4;
  const int kb  = hi * 2;

  // ---------------- Stage 0: obs tile -> LDS (16 KB, padded rows) ----------------
  #pragma unroll
  for (int j4 = 0; j4 < 8; ++j4) {
    const int i4 = tid + 128 * j4;              // 0..1023 float4 slots
    const int m  = i4 >> 6;                     // tile row 0..15
    const int d  = (i4 & 63) * 4;               // column 0..252
    const float4 v = *(const float4*)(obs + ((size_t)(bt0 + m) * NA + n) * DOBS + d);
    *(float4*)(s_obs + m * OBSP + d) = v;
  }
  __syncthreads();

  // ---------------- Step 1: Hobs = obs @ W1[0:256,:] + b1 ----------------
  #pragma unroll
  for (int jj = 0; jj < 2; ++jj) {
    const int jcol = (wid * 2 + jj) * 16 + col;    // 0..127
    v8f acc;
    {
      const float bias = b1[jcol];
      #pragma unroll
      for (int v = 0; v < 8; ++v) acc[v] = bias;
    }
    for (int kk = 0; kk < DOBS; kk += 4) {
      v2f a; a[0] = s_obs[col * OBSP + kk + kb]; a[1] = s_obs[col * OBSP + kk + kb + 1];
      v2f b; b[0] = W1[(kk + kb) * H1D + jcol];  b[1] = W1[(kk + kb + 1) * H1D + jcol];
      acc = wmma4(a, b, acc);
    }
    #pragma unroll
    for (int v = 0; v < 8; ++v) s_hobs[(v + 8 * hi) * H1D + jcol] = acc[v];
  }
  __syncthreads();   // step-1 done: s_obs region may now be overwritten by W2

  // ---------------- Stage 1: weights -> LDS (W1[256:288], W2, W3) ----------------
  #pragma unroll
  for (int j4 = 0; j4 < 8; ++j4) {   // W1 action rows: 4096 floats (contiguous)
    const int j = (tid + 128 * j4) * 4;
    *(float4*)(s_w1a + j) = *(const float4*)(W1 + DOBS * H1D + j);
  }
  #pragma unroll
  for (int j4 = 0; j4 < 16; ++j4) {  // W2: 8192 floats
    const int j = (tid + 128 * j4) * 4;
    *(float4*)(s_w2 + j) = *(const float4*)(W2 + j);
  }
  #pragma unroll
  for (int j4 = 0; j4 < 4; ++j4) {   // W3: 2048 floats
    const int j = (tid + 128 * j4) * 4;
    *(float4*)(s_w3 + j) = *(const float4*)(W3 + j);
  }
  __syncthreads();

  // ---------------- Step 2: 18 variants through layers 1..3 ----------------
  for (int vi = 0; vi < NVAR; ++vi) {
    // A-fragment source for this variant's action slice
    const float* act_row = nullptr;
    if (vi == 0)      act_row = act + ((size_t)(bt0 + col) * NA + n) * AD;
    else if (vi >= 2) act_row = cf  + (((size_t)n * SS + (vi - 2)) * BT + (bt0 + col)) * AD;

    // Prefetch next variant's cf rows (global_prefetch_b8) while we compute
    if (vi + 1 >= 2 && vi + 1 < NVAR) {
      const float* nxt = cf + (((size_t)n * SS + (vi - 1)) * BT + (bt0 + col)) * AD;
      __builtin_prefetch(nxt, 0, 1);
    }

    // Load the 8 A fragments (16x32 action tile) once, reuse for both N-tiles
    v2f av[8];
    if (vi != 1) {
      #pragma unroll
      for (int kq = 0; kq < 8; ++kq) {
        const float2 t = *(const float2*)(act_row + kq * 4 + kb);
        av[kq][0] = t.x; av[kq][1] = t.y;
      }
    }

    // ---- layer 1 action part: h1 = relu(Hobs + act_vi @ W1[256:288,:]) ----
    #pragma unroll
    for (int jj = 0; jj < 2; ++jj) {
      const int jcol = (wid * 2 + jj) * 16 + col;
      v8f acc;
      #pragma unroll
      for (int v = 0; v < 8; ++v) acc[v] = s_hobs[(v + 8 * hi) * H1D + jcol];
      if (vi != 1) {  // zero-action variant contributes nothing here
        #pragma unroll
        for (int kq = 0; kq < 8; ++kq) {
          const int kk = kq * 4;
          v2f b; b[0] = s_w1a[(kk + kb) * H1D + jcol];
                 b[1] = s_w1a[(kk + kb + 1) * H1D + jcol];
          acc = wmma4(av[kq], b, acc);
        }
      }
      #pragma unroll
      for (int v = 0; v < 8; ++v)
        s_h1[(v + 8 * hi) * H1P + jcol] = fmaxf(acc[v], 0.0f);
    }
    __syncthreads();

    // ---- layer 2: h2 = relu(h1 @ W2 + b2), one 16-col tile per wave ----
    {
      const int jcol = wid * 16 + col;   // 0..63
      v8f acc;
      {
        const float bias = b2[jcol];
        #pragma unroll
        for (int v = 0; v < 8; ++v) acc[v] = bias;
      }
      for (int kk = 0; kk < H1D; kk += 4) {
        v2f a; a[0] = s_h1[col * H1P + kk + kb];  a[1] = s_h1[col * H1P + kk + kb + 1];
        v2f b; b[0] = s_w2[(kk + kb) * H2D + jcol]; b[1] = s_w2[(kk + kb + 1) * H2D + jcol];
        acc = wmma4(a, b, acc);
      }
      #pragma unroll
      for (int v = 0; v < 8; ++v)
        s_h2[(v + 8 * hi) * H2P + jcol] = fmaxf(acc[v], 0.0f);
    }
    __syncthreads();

    // ---- layer 3: p = h2 @ W3 + b3, waves 0 and 1 only (2 tiles) ----
    if (wid < 2) {
      const int jcol = wid * 16 + col;   // 0..31
      v8f acc;
      {
        const float bias = b3[jcol];
        #pragma unroll
        for (int v = 0; v < 8; ++v) acc[v] = bias;
      }
      for (int kk = 0; kk < H2D; kk += 4) {
        v2f a; a[0] = s_h2[col * H2P + kk + kb];  a[1] = s_h2[col * H2P + kk + kb + 1];
        v2f b; b[0] = s_w3[(kk + kb) * AD + jcol]; b[1] = s_w3[(kk + kb + 1) * AD + jcol];
        acc = wmma4(a, b, acc);
      }
      if (vi == 0) {
        #pragma unroll
        for (int v = 0; v < 8; ++v) s_pw[(v + 8 * hi) * AD + jcol] = acc[v];
      } else if (vi == 1) {
        #pragma unroll
        for (int v = 0; v < 8; ++v) s_pq[(v + 8 * hi) * AD + jcol] = acc[v];
      } else {
        #pragma unroll
        for (int v = 0; v < 8; ++v) s_pq[(v + 8 * hi) * AD + jcol] += acc[v];
      }
    }
    __syncthreads();
  }

  // ---------------- Step 3: KL(q || p) per row, mean over actions ----------------
  // wave w handles rows 4w..4w+3; lane = action index (32 lanes = 32 actions)
  #pragma unroll
  for (int rr = 0; rr < 4; ++rr) {
    const int r = wid * 4 + rr;
    const float pw = s_pw[r * AD + lane];
    const float pq = s_pq[r * AD + lane] * INV_S1;

    const float mp = wave_max(pw);
    const float mq = wave_max(pq);
    const float ep = expf(pw - mp);
    const float eq = expf(pq - mq);
    const float sp = wave_sum(ep);
    const float sq = wave_sum(eq);

    const float log_p = (pw - mp) - logf(sp);
    const float log_q = (pq - mq) - logf(sq);
    const float q     = eq / sq;
    float kl = q * (log_q - log_p);
    kl = wave_sum(kl);
    if (lane == 0) out[(size_t)(bt0 + r) * NA + n] = kl * (1.0f / AD);
  }
}

extern "C" void kernel_launch(void* const* d_in, const int* in_sizes, int n_in,
                              void* d_out, int out_size, void* d_ws, size_t ws_size,
                              hipStream_t stream) {
  (void)in_sizes; (void)n_in; (void)d_ws; (void)ws_size; (void)out_size;
  const float* obs = (const float*)d_in[0];
  const float* act = (const float*)d_in[1];
  const float* cf  = (const float*)d_in[2];
  const float* W1  = (const float*)d_in[3];
  const float* b1  = (const float*)d_in[4];
  const float* W2  = (const float*)d_in[5];
  const float* b2  = (const float*)d_in[6];
  const float* W3  = (const float*)d_in[7];
  const float* b3  = (const float*)d_in[8];
  float* out = (float*)d_out;

  const int tiles = (NA * BB * TT) / 16;           // 1024 row-tiles of 16
  const size_t lds_bytes = LDS_FLOATS * sizeof(float);  // 82,432 B (gfx1250: up to 320 KB)
  coma_influence_kernel<<<tiles, 128, lds_bytes, stream>>>(
      obs, act, cf, W1, b1, W2, b2, W3, b3, out);
}